// HyperNet_18751827214751
// MI455X (gfx1250) — compile-verified
//
#include <hip/hip_runtime.h>
#include <hip/hip_bf16.h>
#include <math.h>

#define DIMS 64
#define DM1  63
#define NREL 512
#define NC   256      // 1 positive + 255 negative candidates per row
#define STRIDE 68     // LDS row stride (floats): 272 B rows -> 16B aligned, conflict-free WMMA access
#define MSTR   68

typedef __attribute__((ext_vector_type(2))) float v2f;
typedef __attribute__((ext_vector_type(8))) float v8f;

// ---- gfx1250 async global->LDS path (ASYNCcnt) via inline asm ----
// (the clang builtin wants CUDA-LangAS-qualified int4 pointers that HIP source
//  cannot spell, so we emit the instruction directly per CDNA5 ISA)
#define HAVE_ASYNC_LDS 1

#ifdef HAVE_ASYNC_LDS
typedef __attribute__((address_space(3))) void* as3_ptr;

__device__ __forceinline__ void async_copy_b128(const void* g, void* l) {
    unsigned lds_off = (unsigned)(unsigned long long)(as3_ptr)l;   // LDS byte offset
    unsigned long long gaddr = (unsigned long long)g;              // global address
    asm volatile("global_load_async_to_lds_b128 %0, %1, off"
                 :
                 : "v"(lds_off), "v"(gaddr)
                 : "memory");
}
__device__ __forceinline__ void async_wait0() {
    asm volatile("s_wait_asynccnt 0" ::: "memory");
}
#endif

__device__ __forceinline__ float gelu_exact(float x) {
    return 0.5f * x * (1.0f + erff(x * 0.70710678118654752440f));
}

// ---------------------------------------------------------------------------
// Kernel 1: build padded 64x64 rotation matrices M = [[1,0],[0,W]],
// W = Q(v_0) @ ... @ Q(v_62), Q(v) = I - 2 v v^T / (v.v),
// v_k = gelu(rot_w[r, k*63:(k+1)*63]).  grid = 1024 (512 rel x {head,tail}).
// ---------------------------------------------------------------------------
__global__ __launch_bounds__(256) void build_rot_kernel(
    const float* __restrict__ rot_w_head,
    const float* __restrict__ rot_w_tail,
    float* __restrict__ M_out /* [2][NREL][64*64] */)
{
    __shared__ float Iw[DM1 * 65];   // 63 x 65 padded
    __shared__ float vsh[DM1];
    __shared__ float ush[DM1];

    const int tid  = threadIdx.x;
    const int r    = blockIdx.x & (NREL - 1);
    const int side = blockIdx.x >> 9;
    const float* raw = (side ? rot_w_tail : rot_w_head) + (size_t)r * (DM1 * DM1);
    float* Mo = M_out + ((size_t)side * NREL + r) * (DIMS * DIMS);

    for (int idx = tid; idx < DM1 * DM1; idx += 256) {
        int i = idx / DM1, j = idx - i * DM1;
        Iw[i * 65 + j] = (i == j) ? 1.0f : 0.0f;
    }
    __syncthreads();

    for (int k = 0; k < DM1; ++k) {
        if (tid == 0 && k + 1 < DM1)
            __builtin_prefetch(raw + (k + 1) * DM1, 0, 0);   // global_prefetch_b8
        if (tid < DM1) vsh[tid] = gelu_exact(raw[k * DM1 + tid]);
        __syncthreads();
        if (tid < DM1) {
            float vv = 0.f;
            #pragma unroll 7
            for (int j = 0; j < DM1; ++j) vv += vsh[j] * vsh[j];
            float s = 0.f;
            #pragma unroll 7
            for (int j = 0; j < DM1; ++j) s += Iw[tid * 65 + j] * vsh[j];
            ush[tid] = s * (2.0f / vv);
        }
        __syncthreads();
        for (int idx = tid; idx < DM1 * DM1; idx += 256) {
            int i = idx / DM1, j = idx - i * DM1;
            Iw[i * 65 + j] -= ush[i] * vsh[j];
        }
        __syncthreads();
    }

    for (int idx = tid; idx < DIMS * DIMS; idx += 256) {
        int i = idx >> 6, j = idx & 63;
        float val;
        if (i == 0 && j == 0)      val = 1.0f;
        else if (i == 0 || j == 0) val = 0.0f;
        else                       val = Iw[(i - 1) * 65 + (j - 1)];
        Mo[idx] = val;
    }
}

// ---------------------------------------------------------------------------
// Kernel 2: per sample b — async-gather 256 candidate tails into LDS,
// rotate via fp32 WMMA (256x64 @ 64x64), boost, score.
// grid = 1024 blocks, 256 threads = 8 wave32s. Dynamic LDS ~107 KB.
// ---------------------------------------------------------------------------
__global__ __launch_bounds__(256) void score_kernel(
    const int*   __restrict__ pos,
    const int*   __restrict__ neg,
    const float* __restrict__ emb,
    const float* __restrict__ bias_h,
    const float* __restrict__ bias_t,
    const float* __restrict__ boost_h,
    const float* __restrict__ boost_t,
    const float* __restrict__ Mrot,     // [2][NREL][64*64]
    float*       __restrict__ out)      // [B,256]
{
    extern __shared__ float sm[];
    float* Xs   = sm;                        // 256 * 68 (tails; rotated in place)
    float* Mt   = Xs   + NC * STRIDE;        // 64 * 68
    float* Mh   = Mt   + DIMS * MSTR;        // 64 * 68
    float* hraw = Mh   + DIMS * MSTR;        // 64
    float* hrot = hraw + 64;                 // 64
    float* th   = hrot + 64;                 // 64 (boosted head)
    float* roh  = th   + 64;                 // 64 ([0]=0)
    float* rot_ = roh  + 64;                 // 64 ([0]=0)
    float* tbt  = rot_ + 64;                 // 256
    float* misc = tbt  + NC;                 // [0] = tanh(bias_head[u])

    const int tid = threadIdx.x;
    const int b   = blockIdx.x;
    const int u   = pos[b * 3 + 0];
    const int r   = pos[b * 3 + 1];

    const float* MhG = Mrot + (size_t)r * (DIMS * DIMS);
    const float* MtG = Mrot + ((size_t)NREL + r) * (DIMS * DIMS);
    const int cand = (tid == 0) ? pos[b * 3 + 2] : neg[b * 255 + tid - 1];
    const float* csrc = emb + (size_t)cand * DIMS;

#ifdef HAVE_ASYNC_LDS
    {   // rotation matrices: 1024 16B chunks each, 4 per thread
        #pragma unroll
        for (int i2 = 0; i2 < 4; ++i2) {
            int c   = tid + 256 * i2;
            int row = c >> 4, q = c & 15;
            async_copy_b128(MhG + row * 64 + q * 4, Mh + row * MSTR + q * 4);
            async_copy_b128(MtG + row * 64 + q * 4, Mt + row * MSTR + q * 4);
        }
        // candidate gather: one 256B row per thread, L2 -> LDS, no VGPR round-trip
        float* dst = Xs + tid * STRIDE;
        #pragma unroll
        for (int q = 0; q < 16; ++q)
            async_copy_b128(csrc + q * 4, dst + q * 4);
    }
#else
    for (int idx = tid; idx < DIMS * DIMS; idx += 256) {
        int i = idx >> 6, j = idx & 63;
        Mh[i * MSTR + j] = MhG[idx];
        Mt[i * MSTR + j] = MtG[idx];
    }
    {
        const float4* row = (const float4*)csrc;
        float* dst = Xs + tid * STRIDE;
        #pragma unroll
        for (int q = 0; q < 16; ++q) {
            float4 v = row[q];
            dst[q * 4 + 0] = v.x; dst[q * 4 + 1] = v.y;
            dst[q * 4 + 2] = v.z; dst[q * 4 + 3] = v.w;
        }
    }
#endif
    // scalar work overlapped with the async gather
    tbt[tid] = tanhf(bias_t[cand]);
    if (tid < DIMS) hraw[tid] = emb[(size_t)u * DIMS + tid];
    if (tid < DM1) {
        roh[tid + 1]  = tanhf(boost_h[r * DM1 + tid]) * (1.0f / 64.0f);
        rot_[tid + 1] = tanhf(boost_t[r * DM1 + tid]) * (1.0f / 64.0f);
    }
    if (tid == 63) { roh[0] = 0.f; rot_[0] = 0.f; misc[0] = tanhf(bias_h[u]); }
#ifdef HAVE_ASYNC_LDS
    async_wait0();
#endif
    __syncthreads();

    // ---- head rotation (tiny: plain VALU) ----
    if (tid < DIMS) {
        float s = 0.f;
        #pragma unroll 8
        for (int k = 0; k < DIMS; ++k) s += hraw[k] * Mh[k * MSTR + tid];
        hrot[tid] = s;
    }
    __syncthreads();

    // ---- head boost ----
    if (tid < DIMS) {
        float v2 = 0.f, dot = 0.f;
        #pragma unroll 7
        for (int j = 1; j < DIMS; ++j) { v2 += roh[j] * roh[j]; dot += hrot[j] * roh[j]; }
        float zeta = 1.0f / (sqrtf(1.0f - v2) + 1e-8f);
        float t0 = hrot[0];
        if (tid == 0) th[0] = zeta * t0 - zeta * dot;
        else th[tid] = -zeta * t0 * roh[tid] + hrot[tid]
                       + (zeta - 1.0f) / (v2 + 1e-9f) * roh[tid] * dot;
    }

    // ---- tail rotation: Y = Xs(256x64) @ Mt(64x64), fp32 WMMA.
    //      Both 16-row M-blocks of a wave share one k-loop so each set of
    //      4 B-fragments feeds 8 WMMAs; in-place writeback (wave-private rows).
    {
        const int lane = tid & 31;
        const int wave = tid >> 5;
        const int half = lane >> 4;
        const int l16  = lane & 15;
        const float* ar0 = Xs + (wave * 16 + l16) * STRIDE;        // M-block wave
        const float* ar1 = Xs + ((wave + 8) * 16 + l16) * STRIDE;  // M-block wave+8
        v8f c00 = {}, c01 = {}, c02 = {}, c03 = {};
        v8f c10 = {}, c11 = {}, c12 = {}, c13 = {};
        for (int k0 = 0; k0 < DIMS; k0 += 4) {
            const int ka = k0 + half * 2;
            v2f a0, a1, b0, b1, b2, b3;
            a0.x = ar0[ka]; a0.y = ar0[ka + 1];
            a1.x = ar1[ka]; a1.y = ar1[ka + 1];
            const float* m0 = Mt + ka * MSTR + l16;
            const float* m1 = Mt + (ka + 1) * MSTR + l16;
            b0.x = m0[ 0]; b0.y = m1[ 0];
            b1.x = m0[16]; b1.y = m1[16];
            b2.x = m0[32]; b2.y = m1[32];
            b3.x = m0[48]; b3.y = m1[48];
            c00 = __builtin_amdgcn_wmma_f32_16x16x4_f32(false, a0, false, b0, (short)0, c00, false, false);
            c10 = __builtin_amdgcn_wmma_f32_16x16x4_f32(false, a1, false, b0, (short)0, c10, false, false);
            c01 = __builtin_amdgcn_wmma_f32_16x16x4_f32(false, a0, false, b1, (short)0, c01, false, false);
            c11 = __builtin_amdgcn_wmma_f32_16x16x4_f32(false, a1, false, b1, (short)0, c11, false, false);
            c02 = __builtin_amdgcn_wmma_f32_16x16x4_f32(false, a0, false, b2, (short)0, c02, false, false);
            c12 = __builtin_amdgcn_wmma_f32_16x16x4_f32(false, a1, false, b2, (short)0, c12, false, false);
            c03 = __builtin_amdgcn_wmma_f32_16x16x4_f32(false, a0, false, b3, (short)0, c03, false, false);
            c13 = __builtin_amdgcn_wmma_f32_16x16x4_f32(false, a1, false, b3, (short)0, c13, false, false);
        }
        // C/D layout: VGPR g -> row (mb*16 + g + half*8), col (nt*16 + l16)
        #pragma unroll
        for (int g = 0; g < 8; ++g) {
            int r0 = (wave * 16 + g + half * 8) * STRIDE + l16;
            int r1 = ((wave + 8) * 16 + g + half * 8) * STRIDE + l16;
            Xs[r0 +  0] = c00[g];  Xs[r0 + 16] = c01[g];
            Xs[r0 + 32] = c02[g];  Xs[r0 + 48] = c03[g];
            Xs[r1 +  0] = c10[g];  Xs[r1 + 16] = c11[g];
            Xs[r1 + 32] = c12[g];  Xs[r1 + 48] = c13[g];
        }
    }
    __syncthreads();

    // ---- tail boost + squared-Lorentz score, one candidate per thread ----
    {
        float v2 = 0.f;
        #pragma unroll 7
        for (int j = 1; j < DIMS; ++j) v2 += rot_[j] * rot_[j];
        float zeta = 1.0f / (sqrtf(1.0f - v2) + 1e-8f);
        const float* x = Xs + tid * STRIDE;
        float t0  = x[0];
        float dot = 0.f;
        #pragma unroll 7
        for (int j = 1; j < DIMS; ++j) dot += x[j] * rot_[j];
        float tt0  = zeta * t0 - zeta * dot;
        float dt0  = tt0 - th[0];
        float coef = (zeta - 1.0f) / (v2 + 1e-9f) * dot;
        float acc  = 0.f;
        #pragma unroll 7
        for (int j = 1; j < DIMS; ++j) {
            float ttj = -zeta * t0 * rot_[j] + x[j] + coef * rot_[j];
            float d   = ttj - th[j];
            acc += d * d;
        }
        float mkv = -dt0 * dt0 + acc;
        out[(size_t)b * NC + tid] = 0.85f - mkv + misc[0] + tbt[tid];
    }
}

extern "C" void kernel_launch(void* const* d_in, const int* in_sizes, int n_in,
                              void* d_out, int out_size, void* d_ws, size_t ws_size,
                              hipStream_t stream) {
    const int*   pos    = (const int*)  d_in[0];
    const int*   neg    = (const int*)  d_in[1];
    const float* emb    = (const float*)d_in[2];
    const float* bias_h = (const float*)d_in[3];
    const float* bias_t = (const float*)d_in[4];
    const float* rwh    = (const float*)d_in[5];
    const float* bwh    = (const float*)d_in[6];
    const float* rwt    = (const float*)d_in[7];
    const float* bwt    = (const float*)d_in[8];
    float* outp = (float*)d_out;
    float* Mrot = (float*)d_ws;   // 2 * 512 * 64 * 64 floats = 16 MB

    build_rot_kernel<<<2 * NREL, 256, 0, stream>>>(rwh, rwt, Mrot);

    size_t smem = (size_t)(NC * STRIDE + 2 * DIMS * MSTR + 64 * 5 + NC + 8) * sizeof(float);
    score_kernel<<<1024, 256, smem, stream>>>(pos, neg, emb, bias_h, bias_t,
                                              bwh, bwt, Mrot, outp);
}